// MarketingGNN_71004399338031
// MI455X (gfx1250) — compile-verified
//
#include <hip/hip_runtime.h>

// ---------------------------------------------------------------------------
// MarketingGNN product head for MI455X (gfx1250, wave32).
// Only the product branch of the HeteroConv matters (reference discards
// demo/platform outputs), so we compute:
//   acc2/cnt2 = segment-sum/count of x_demo over ei_rev_targets  (2M edges)
//   acc5/cnt5 = segment-sum/count of x_prod over ei_self        (200k edges)
//   h  = leaky_relu(0.5*(mean2@Wl2 + mean5@Wl5 + x_p@(Wr2+Wr5) + bl2+bl5))
//   out = h @ W_out + b_out                                     [200000 x 8]
// Dense math uses V_WMMA_F32_16X16X4_F32 (full fp32 fidelity; GEMMs are far
// from the bottleneck -- the L2-resident atomic scatter dominates).
// ---------------------------------------------------------------------------

typedef __attribute__((ext_vector_type(2))) float v2f;
typedef __attribute__((ext_vector_type(8))) float v8f;

__global__ void zero_ws_kernel(float* __restrict__ p, int n) {
    int i = blockIdx.x * blockDim.x + threadIdx.x;
    int stride = gridDim.x * blockDim.x;
    for (; i < n; i += stride) p[i] = 0.0f;
}

// One thread per (edge, feature-component). Lanes 0..31 of a wave cover the
// 32 components of one edge -> coalesced 128B gather from x_src row and
// coalesced global_atomic_add_f32 into the accumulator row (L2-resident).
__global__ void scatter_edges_kernel(const float* __restrict__ xsrc,
                                     const int* __restrict__ ei, int E,
                                     float* __restrict__ acc,
                                     float* __restrict__ cnt) {
    long long gid = (long long)blockIdx.x * blockDim.x + threadIdx.x;
    if (gid >= (long long)E * 32) return;
    int e = (int)(gid >> 5);
    int c = (int)(gid & 31);
    int src = ei[e];         // row 0 of [2,E] edge_index
    int dst = ei[E + e];     // row 1
    atomicAdd(acc + (long long)dst * 32 + c, xsrc[(long long)src * 32 + c]);
    if (c == 0) atomicAdd(cnt + dst, 1.0f);
}

// 128 threads = 4 waves; each wave owns one 16-product tile.
// Stage 1: D[16,64] = mean2@Wl2 + mean5@Wl5 + x_p@(Wr2+Wr5) + (bl2+bl5)
//          via 96 V_WMMA_F32_16X16X4_F32 (4 N-tiles x 8 K-steps x 3 terms).
// Stage 2: activated tile -> LDS (transpose C/D layout to A layout),
//          then 16 WMMAs for the 64->8 head (N padded to 16 with zeros).
__global__ __launch_bounds__(128)
void product_head_kernel(const float* __restrict__ xp,
                         const float* __restrict__ acc2, const float* __restrict__ cnt2,
                         const float* __restrict__ acc5, const float* __restrict__ cnt5,
                         const float* __restrict__ Wl2, const float* __restrict__ bl2,
                         const float* __restrict__ Wr2,
                         const float* __restrict__ Wl5, const float* __restrict__ bl5,
                         const float* __restrict__ Wr5,
                         const float* __restrict__ Wout, const float* __restrict__ bout,
                         float* __restrict__ out, int nprod) {
    __shared__ float lds_h[4][16 * 64];

    const int lane  = threadIdx.x & 31;
    const int wave  = threadIdx.x >> 5;
    const int tile  = blockIdx.x * 4 + wave;
    const int r0    = tile * 16;
    const int col16 = lane & 15;              // row within tile == N within N-tile
    const int kk    = (lane < 16) ? 0 : 2;    // K split across lane halves
    int r = r0 + col16;                       // product row this lane loads (A side)
    int rclamp = (r < nprod) ? r : (nprod - 1);

    const float ic2 = 1.0f / fmaxf(cnt2[rclamp], 1.0f);
    const float ic5 = 1.0f / fmaxf(cnt5[rclamp], 1.0f);

    const float* a2base = acc2 + (long long)rclamp * 32;
    const float* a5base = acc5 + (long long)rclamp * 32;
    const float* xpbase = xp   + (long long)rclamp * 32;

    // C init = combined bias (depends only on N = col16, broadcast over M).
    v8f cacc[4];
#pragma unroll
    for (int nt = 0; nt < 4; ++nt) {
        float b = bl2[16 * nt + col16] + bl5[16 * nt + col16];
#pragma unroll
        for (int v = 0; v < 8; ++v) cacc[nt][v] = b;
    }

#pragma unroll
    for (int ks = 0; ks < 8; ++ks) {
        const int kb = 4 * ks + kk;
        v2f am2 = { a2base[kb] * ic2, a2base[kb + 1] * ic2 };
        v2f am5 = { a5base[kb] * ic5, a5base[kb + 1] * ic5 };
        v2f axp = { xpbase[kb],        xpbase[kb + 1] };
#pragma unroll
        for (int nt = 0; nt < 4; ++nt) {
            const int wc = 16 * nt + col16;
            v2f b2 = { Wl2[kb * 64 + wc],  Wl2[(kb + 1) * 64 + wc] };
            v2f b5 = { Wl5[kb * 64 + wc],  Wl5[(kb + 1) * 64 + wc] };
            v2f br = { Wr2[kb * 64 + wc] + Wr5[kb * 64 + wc],
                       Wr2[(kb + 1) * 64 + wc] + Wr5[(kb + 1) * 64 + wc] };
            cacc[nt] = __builtin_amdgcn_wmma_f32_16x16x4_f32(
                false, am2, false, b2, (short)0, cacc[nt], false, false);
            cacc[nt] = __builtin_amdgcn_wmma_f32_16x16x4_f32(
                false, am5, false, b5, (short)0, cacc[nt], false, false);
            cacc[nt] = __builtin_amdgcn_wmma_f32_16x16x4_f32(
                false, axp, false, br, (short)0, cacc[nt], false, false);
        }
    }

    // Activation + C/D-layout -> row-major LDS (transpose for stage-2 A reads).
#pragma unroll
    for (int nt = 0; nt < 4; ++nt) {
#pragma unroll
        for (int v = 0; v < 8; ++v) {
            float p = 0.5f * cacc[nt][v];
            float h = (p >= 0.0f) ? p : 0.01f * p;   // leaky_relu slope 0.01
            int m = (lane < 16) ? v : (v + 8);        // M index of this element
            lds_h[wave][m * 64 + 16 * nt + col16] = h;
        }
    }
    __syncthreads();

    // Stage 2: out[16,8] = h[16,64] @ W_out[64,8] + b_out (N padded to 16).
    v8f d;
    {
        float ob = (col16 < 8) ? bout[col16] : 0.0f;
#pragma unroll
        for (int v = 0; v < 8; ++v) d[v] = ob;
    }
    const float* hrow = &lds_h[wave][col16 * 64];
#pragma unroll
    for (int ks = 0; ks < 16; ++ks) {
        const int kb = 4 * ks + kk;
        v2f a = { hrow[kb], hrow[kb + 1] };
        v2f b = { (col16 < 8) ? Wout[kb * 8 + col16] : 0.0f,
                  (col16 < 8) ? Wout[(kb + 1) * 8 + col16] : 0.0f };
        d = __builtin_amdgcn_wmma_f32_16x16x4_f32(
            false, a, false, b, (short)0, d, false, false);
    }

    if (col16 < 8) {
#pragma unroll
        for (int v = 0; v < 8; ++v) {
            int m  = (lane < 16) ? v : (v + 8);
            int rr = r0 + m;
            if (rr < nprod) out[(long long)rr * 8 + col16] = d[v];
        }
    }
}

extern "C" void kernel_launch(void* const* d_in, const int* in_sizes, int n_in,
                              void* d_out, int out_size, void* d_ws, size_t ws_size,
                              hipStream_t stream) {
    const float* x_product     = (const float*)d_in[0];
    const float* x_demographic = (const float*)d_in[1];
    // d_in[2] x_platform, d_in[3] ei_targets, d_in[5] ei_uses, d_in[6] ei_rev_uses: unused.
    const int* ei_rev_targets  = (const int*)d_in[4];
    const int* ei_self         = (const int*)d_in[7];
    const float* Wl2 = (const float*)d_in[11];
    const float* bl2 = (const float*)d_in[12];
    const float* Wr2 = (const float*)d_in[13];
    const float* Wl5 = (const float*)d_in[20];
    const float* bl5 = (const float*)d_in[21];
    const float* Wr5 = (const float*)d_in[22];
    const float* Wout = (const float*)d_in[23];
    const float* bout = (const float*)d_in[24];
    float* out = (float*)d_out;

    const int nprod = in_sizes[0] / 32;
    const int E_T   = in_sizes[4] / 2;
    const int E_S   = in_sizes[7] / 2;

    float* acc2 = (float*)d_ws;
    float* cnt2 = acc2 + (size_t)nprod * 32;
    float* acc5 = cnt2 + nprod;
    float* cnt5 = acc5 + (size_t)nprod * 32;
    const int totalF = nprod * 66;  // acc2+cnt2+acc5+cnt5

    // 1) zero accumulators (every call: harness does not re-poison / restore).
    zero_ws_kernel<<<4096, 256, 0, stream>>>((float*)d_ws, totalF);

    // 2) edge scatters (atomic segment-sum + degree count).
    {
        long long th = (long long)E_T * 32;
        scatter_edges_kernel<<<(unsigned)((th + 255) / 256), 256, 0, stream>>>(
            x_demographic, ei_rev_targets, E_T, acc2, cnt2);
    }
    {
        long long th = (long long)E_S * 32;
        scatter_edges_kernel<<<(unsigned)((th + 255) / 256), 256, 0, stream>>>(
            x_product, ei_self, E_S, acc5, cnt5);
    }

    // 3) fused mean + GEMMs + leaky_relu + output head (WMMA f32).
    const int blocks = (nprod + 63) / 64;  // 4 waves/block * 16 rows/wave
    product_head_kernel<<<blocks, 128, 0, stream>>>(
        x_product, acc2, cnt2, acc5, cnt5,
        Wl2, bl2, Wr2, Wl5, bl5, Wr5, Wout, bout, out, nprod);
}